// DIRAC_2482491097661
// MI455X (gfx1250) — compile-verified
//
#include <hip/hip_runtime.h>

typedef __attribute__((ext_vector_type(2))) float v2f;
typedef __attribute__((ext_vector_type(8))) float v8f;

#define TILE_ROWS 128

#if __has_builtin(__builtin_amdgcn_global_load_async_to_lds_b32)
#define HAS_ASYNC_LDS 1
#else
#define HAS_ASYNC_LDS 0
#endif

typedef __attribute__((address_space(3))) int  lds_i32;
typedef __attribute__((address_space(1))) int  glb_i32;

// Generic LDS pointer -> addrspace(3) pointer. The generic form of an LDS
// address keeps the DS byte offset in bits [31:0] (ISA 10.2 aperture rules),
// so truncation yields the AS3 pointer. Builtin wants int-typed pointers.
__device__ inline lds_i32* to_lds(float* p) {
    return (lds_i32*)(unsigned)(uintptr_t)p;
}
__device__ inline glb_i32* to_glb(const float* p) {
    return (glb_i32*)(uintptr_t)p;
}

__device__ inline void wait_async_lds() {
#if HAS_ASYNC_LDS
#if __has_builtin(__builtin_amdgcn_s_wait_asynccnt)
    __builtin_amdgcn_s_wait_asynccnt(0);
#else
    asm volatile("s_wait_asynccnt 0x0" ::: "memory");
#endif
#endif
}

// ---------------------------------------------------------------------------
// Generic GNN layer:  out = pool3?(relu( [A_x | A_e] @ blockdiag(Wx,We) + [bx|be] ))
//   EDGE_MODE: A_x row = x[src[e]] + x[dst[e]] (gather, VGPR path)
//              A_e row = ein[e]                (contiguous slab, async->LDS)
//   NODE_MODE: A_x row = x[n]                  (contiguous slab, async->LDS)
//              A_e row = agg[n]                (contiguous slab, async->LDS)
// 256 threads = 8 waves; each wave computes one 16-row tile with
// V_WMMA_F32_16X16X4_F32 (fp32-exact, matches reference numerics).
// ---------------------------------------------------------------------------
template<int CX, int CE, int CXO, int CEO, bool POOL, bool EDGE_MODE>
__global__ __launch_bounds__(256) void gnn_layer_kernel(
    const float* __restrict__ xin, const float* __restrict__ ein,
    const int* __restrict__ src, const int* __restrict__ dst,
    const float* __restrict__ Wx, const float* __restrict__ bx,
    const float* __restrict__ We, const float* __restrict__ be,
    float* __restrict__ out, int nrows)
{
    constexpr int K   = CX + CE;
    constexpr int KP  = (K + 3) & ~3;          // K padded to multiple of 4
    constexpr int CO  = CXO + CEO;
    constexpr int CT  = (CO + 15) / 16;        // 16-wide column tiles
    constexpr int COP = CT * 16;
    constexpr int OC  = POOL ? (CO / 3) : CO;  // output channels after pool

    __shared__ float sB[KP][COP];              // block-diagonal weights
    __shared__ float sBias[COP];
    __shared__ float sAx[TILE_ROWS * CX];      // x-part panel (row-major)
    __shared__ float sEA[TILE_ROWS * CE];      // edge/agg-part panel (row-major slab)
    __shared__ float sC[TILE_ROWS][COP];       // matmul result panel

    const int tid  = threadIdx.x;
    const int base = blockIdx.x * TILE_ROWS;

#if HAS_ASYNC_LDS
    // Kick off async global->LDS slab copies first (tracked by ASYNCcnt).
    {
        const long long eaBase = (long long)base * CE;
        const long long eaTot  = (long long)nrows * CE;
        for (int idx = tid; idx < TILE_ROWS * CE; idx += 256) {
            if (eaBase + idx < eaTot)
                __builtin_amdgcn_global_load_async_to_lds_b32(
                    to_glb(ein) + (eaBase + idx), to_lds(sEA) + idx, 0, 0);
        }
        if (!EDGE_MODE) {
            const long long xBase = (long long)base * CX;
            const long long xTot  = (long long)nrows * CX;
            for (int idx = tid; idx < TILE_ROWS * CX; idx += 256) {
                if (xBase + idx < xTot)
                    __builtin_amdgcn_global_load_async_to_lds_b32(
                        to_glb(xin) + (xBase + idx), to_lds(sAx) + idx, 0, 0);
            }
        }
    }
#else
    {
        const long long eaBase = (long long)base * CE;
        const long long eaTot  = (long long)nrows * CE;
        for (int idx = tid; idx < TILE_ROWS * CE; idx += 256)
            sEA[idx] = (eaBase + idx < eaTot) ? ein[eaBase + idx] : 0.0f;
        if (!EDGE_MODE) {
            const long long xBase = (long long)base * CX;
            const long long xTot  = (long long)nrows * CX;
            for (int idx = tid; idx < TILE_ROWS * CX; idx += 256)
                sAx[idx] = (xBase + idx < xTot) ? xin[xBase + idx] : 0.0f;
        }
    }
#endif

    // Stage block-diagonal weight matrix + concatenated bias in LDS.
    for (int idx = tid; idx < KP * COP; idx += 256) {
        int k = idx / COP, c = idx % COP;
        float v = 0.0f;
        if (k < CX && c < CXO)                              v = Wx[k * CXO + c];
        else if (k >= CX && k < K && c >= CXO && c < CO)    v = We[(k - CX) * CEO + (c - CXO)];
        sB[k][c] = v;
    }
    for (int c = tid; c < COP; c += 256)
        sBias[c] = (c < CXO) ? bx[c] : ((c < CO) ? be[c - CXO] : 0.0f);

    // Gathered x-part for edge layers (divergent addresses -> VGPR path).
    if (EDGE_MODE) {
        for (int idx = tid; idx < TILE_ROWS * CX; idx += 256) {
            int r = idx / CX, k = idx % CX;
            int row = base + r;
            float v = 0.0f;
            if (row < nrows)
                v = xin[src[row] * CX + k] + xin[dst[row] * CX + k];
            sAx[idx] = v;
        }
    }

    wait_async_lds();
    __syncthreads();

    // WMMA: wave w owns rows [16w, 16w+16).
    {
        const int wave = tid >> 5;
        const int lane = tid & 31;
        const int half = lane >> 4;    // lanes 16..31 hold the K+2 slice (ISA 7.12.2)
        const int lrow = lane & 15;    // M index in both halves
        const int rt   = wave * 16;

        auto loadA = [&](int r, int k) -> float {
            float v = 0.0f;
            if (k < CX)      v = sAx[r * CX + k];
            else if (k < K)  v = sEA[r * CE + (k - CX)];
            return v;
        };

        for (int ct = 0; ct < CT; ++ct) {
            v8f c = {};
            #pragma unroll
            for (int k0 = 0; k0 < KP; k0 += 4) {
                v2f a, b;
                a.x = loadA(rt + lrow, k0 + 2 * half);
                a.y = loadA(rt + lrow, k0 + 2 * half + 1);
                b.x = sB[k0 + 2 * half][ct * 16 + lrow];
                b.y = sB[k0 + 2 * half + 1][ct * 16 + lrow];
                c = __builtin_amdgcn_wmma_f32_16x16x4_f32(
                        false, a, false, b, (short)0, c, false, false);
            }
            // C layout: VGPR j -> row rt + j + 8*half, col ct*16 + lrow
            #pragma unroll
            for (int j = 0; j < 8; ++j)
                sC[rt + j + 8 * half][ct * 16 + lrow] = c[j];
        }
    }
    __syncthreads();

    // Epilogue: bias + relu (+ maxpool3 across the full concat), store.
    for (int idx = tid; idx < TILE_ROWS * OC; idx += 256) {
        int r = idx / OC, oc = idx % OC;
        int row = base + r;
        if (row < nrows) {
            float v;
            if (POOL) {
                float v0 = sC[r][oc * 3 + 0] + sBias[oc * 3 + 0];
                float v1 = sC[r][oc * 3 + 1] + sBias[oc * 3 + 1];
                float v2 = sC[r][oc * 3 + 2] + sBias[oc * 3 + 2];
                v = fmaxf(fmaxf(v0, v1), v2);
            } else {
                v = sC[r][oc] + sBias[oc];
            }
            out[row * OC + oc] = fmaxf(v, 0.0f);
        }
    }
}

// ---------------------------------------------------------------------------
__global__ __launch_bounds__(256) void zero_kernel(float* __restrict__ p, int n)
{
    int i = blockIdx.x * 256 + threadIdx.x;
    if (i < n) p[i] = 0.0f;
}

// segment_sum(ea, src): one thread per (edge, channel), fp32 L2 atomics.
template<int CE>
__global__ __launch_bounds__(256) void scatter_add_kernel(
    const float* __restrict__ ea, const int* __restrict__ src,
    float* __restrict__ agg, int nE)
{
    long long idx = (long long)blockIdx.x * 256 + threadIdx.x;
    long long tot = (long long)nE * CE;
    if (idx < tot) {
        int e  = (int)(idx / CE);
        int ch = (int)(idx - (long long)e * CE);
        atomicAdd(&agg[src[e] * CE + ch], ea[idx]);
    }
}

// column sum of (n,6) -> sums[6]
__global__ __launch_bounds__(256) void colsum6_kernel(
    const float* __restrict__ x, float* __restrict__ sums, int n)
{
    __shared__ float s[256][6];
    float acc[6] = {0.f, 0.f, 0.f, 0.f, 0.f, 0.f};
    for (int r = blockIdx.x * 256 + threadIdx.x; r < n; r += gridDim.x * 256) {
        #pragma unroll
        for (int c = 0; c < 6; ++c) acc[c] += x[r * 6 + c];
    }
    #pragma unroll
    for (int c = 0; c < 6; ++c) s[threadIdx.x][c] = acc[c];
    __syncthreads();
    for (int off = 128; off > 0; off >>= 1) {
        if ((int)threadIdx.x < off) {
            #pragma unroll
            for (int c = 0; c < 6; ++c) s[threadIdx.x][c] += s[threadIdx.x + off][c];
        }
        __syncthreads();
    }
    if (threadIdx.x < 6) atomicAdd(&sums[threadIdx.x], s[0][threadIdx.x]);
}

// Fused MLP head: q=[state|x5] -> fc1(12x100) relu -> fc2(100x25) relu -> fc3(25x1) relu
__global__ __launch_bounds__(256) void fc_kernel(
    const float* __restrict__ x5, const float* __restrict__ sums,
    const float* __restrict__ w1, const float* __restrict__ b1,
    const float* __restrict__ w2, const float* __restrict__ b2,
    const float* __restrict__ w3, const float* __restrict__ b3,
    float* __restrict__ out, int n)
{
    __shared__ float sw1[12 * 100];
    __shared__ float sb1[100];
    __shared__ float sw2[100 * 25];
    __shared__ float sb2[25];
    __shared__ float sw3[25];
    __shared__ float sbias3;
    __shared__ float sstate[6];

    const int tid = threadIdx.x;
    for (int i = tid; i < 1200; i += 256) sw1[i] = w1[i];
    for (int i = tid; i < 100;  i += 256) sb1[i] = b1[i];
    for (int i = tid; i < 2500; i += 256) sw2[i] = w2[i];
    for (int i = tid; i < 25;   i += 256) sb2[i] = b2[i];
    for (int i = tid; i < 25;   i += 256) sw3[i] = w3[i];
    if (tid == 0) sbias3 = b3[0];
    if (tid < 6) sstate[tid] = sums[tid];
    __syncthreads();

    int i = blockIdx.x * 256 + tid;
    if (i >= n) return;

    float q[12];
    #pragma unroll
    for (int c = 0; c < 6; ++c) { q[c] = sstate[c]; q[6 + c] = x5[i * 6 + c]; }

    float h1[100];
    #pragma unroll 4
    for (int j = 0; j < 100; ++j) {
        float acc = sb1[j];
        #pragma unroll
        for (int k = 0; k < 12; ++k) acc += q[k] * sw1[k * 100 + j];
        h1[j] = fmaxf(acc, 0.0f);
    }
    float oacc = sbias3;
    for (int j = 0; j < 25; ++j) {
        float acc = sb2[j];
        #pragma unroll 4
        for (int k = 0; k < 100; ++k) acc += h1[k] * sw2[k * 25 + j];
        oacc += fmaxf(acc, 0.0f) * sw3[j];
    }
    out[i] = fmaxf(oacc, 0.0f);
}

// ---------------------------------------------------------------------------
extern "C" void kernel_launch(void* const* d_in, const int* in_sizes, int n_in,
                              void* d_out, int out_size, void* d_ws, size_t ws_size,
                              hipStream_t stream)
{
    (void)n_in; (void)out_size; (void)ws_size;

    const float* x0  = (const float*)d_in[0];
    const int*   ei  = (const int*)d_in[1];
    const float* ea0 = (const float*)d_in[2];
    const int N = in_sizes[0] / 5;
    const int E = in_sizes[1] / 2;
    const int* src = ei;
    const int* dst = ei + E;

    // params flattened in _DIMS order: (w,b) pairs starting at d_in[3]
    auto W  = [&](int i) { return (const float*)d_in[3 + 2 * i]; };
    auto Bv = [&](int i) { return (const float*)d_in[4 + 2 * i]; };

    // workspace carve-out (floats)
    float* ws   = (float*)d_ws;
    float* eaA  = ws;                          // E*10
    float* eaB  = eaA + (size_t)E * 10;        // E*10
    float* xA   = eaB + (size_t)E * 10;        // N*10
    float* xB   = xA  + (size_t)N * 10;        // N*10
    float* agg  = xB  + (size_t)N * 10;        // N*10
    float* sums = agg + (size_t)N * 10;        // 6

    const int ebl = (E + TILE_ROWS - 1) / TILE_ROWS;
    const int nbl = (N + TILE_ROWS - 1) / TILE_ROWS;
    auto g = [](long long n) { return (unsigned)((n + 255) / 256); };

    // ---- layer 1 ----
    gnn_layer_kernel<5, 1, 10, 2, true, true><<<ebl, 256, 0, stream>>>(
        x0, ea0, src, dst, W(0), Bv(0), W(1), Bv(1), eaA, E);              // ea1 (E,4)
    zero_kernel<<<g((long long)N * 4), 256, 0, stream>>>(agg, N * 4);
    scatter_add_kernel<4><<<g((long long)E * 4), 256, 0, stream>>>(eaA, src, agg, E);
    gnn_layer_kernel<5, 4, 18, 12, true, false><<<nbl, 256, 0, stream>>>(
        x0, agg, src, dst, W(2), Bv(2), W(3), Bv(3), xA, N);               // x1 (N,10)

    // ---- layer 2 ----
    gnn_layer_kernel<10, 4, 18, 12, true, true><<<ebl, 256, 0, stream>>>(
        xA, eaA, src, dst, W(4), Bv(4), W(5), Bv(5), eaB, E);              // ea2 (E,10)
    zero_kernel<<<g((long long)N * 10), 256, 0, stream>>>(agg, N * 10);
    scatter_add_kernel<10><<<g((long long)E * 10), 256, 0, stream>>>(eaB, src, agg, E);
    gnn_layer_kernel<10, 10, 18, 12, true, false><<<nbl, 256, 0, stream>>>(
        xA, agg, src, dst, W(6), Bv(6), W(7), Bv(7), xB, N);               // x2 (N,10)

    // ---- layer 3 ----
    gnn_layer_kernel<10, 10, 18, 12, true, true><<<ebl, 256, 0, stream>>>(
        xB, eaB, src, dst, W(8), Bv(8), W(9), Bv(9), eaA, E);              // ea3 (E,10)
    zero_kernel<<<g((long long)N * 10), 256, 0, stream>>>(agg, N * 10);
    scatter_add_kernel<10><<<g((long long)E * 10), 256, 0, stream>>>(eaA, src, agg, E);
    gnn_layer_kernel<10, 10, 18, 12, true, false><<<nbl, 256, 0, stream>>>(
        xB, agg, src, dst, W(10), Bv(10), W(11), Bv(11), xA, N);           // x3 (N,10)

    // ---- layer 4 ----
    gnn_layer_kernel<10, 10, 18, 12, true, true><<<ebl, 256, 0, stream>>>(
        xA, eaA, src, dst, W(12), Bv(12), W(13), Bv(13), eaB, E);          // ea4 (E,10)
    zero_kernel<<<g((long long)N * 10), 256, 0, stream>>>(agg, N * 10);
    scatter_add_kernel<10><<<g((long long)E * 10), 256, 0, stream>>>(eaB, src, agg, E);
    gnn_layer_kernel<10, 10, 18, 12, true, false><<<nbl, 256, 0, stream>>>(
        xA, agg, src, dst, W(14), Bv(14), W(15), Bv(15), xB, N);           // x4 (N,10)

    // ---- layer 5 (no pool) ----
    gnn_layer_kernel<10, 10, 3, 3, false, true><<<ebl, 256, 0, stream>>>(
        xB, eaB, src, dst, W(16), Bv(16), W(17), Bv(17), eaA, E);          // ea5 (E,6)
    zero_kernel<<<g((long long)N * 6), 256, 0, stream>>>(agg, N * 6);
    scatter_add_kernel<6><<<g((long long)E * 6), 256, 0, stream>>>(eaA, src, agg, E);
    gnn_layer_kernel<10, 6, 3, 3, false, false><<<nbl, 256, 0, stream>>>(
        xB, agg, src, dst, W(18), Bv(18), W(19), Bv(19), xA, N);           // x5 (N,6)

    // ---- readout + MLP head ----
    zero_kernel<<<1, 256, 0, stream>>>(sums, 6);
    colsum6_kernel<<<256, 256, 0, stream>>>(xA, sums, N);
    fc_kernel<<<g(N), 256, 0, stream>>>(
        xA, sums, W(20), Bv(20), W(21), Bv(21), W(22), Bv(22), (float*)d_out, N);
}